// FewTPLinkerPlus_78950088835209
// MI455X (gfx1250) — compile-verified
//
#include <hip/hip_runtime.h>
#include <math.h>

typedef __attribute__((ext_vector_type(2))) float v2f;
typedef __attribute__((ext_vector_type(8))) float v8f;

#define EP   8
#define SUP  20
#define QRY  20
#define LSEQ 128
#define DHID 1024
#define MHID 256
#define QL   (QRY * LSEQ)      /* 2560 */
#define RSUP (EP * SUP * LSEQ) /* 20480 */

// ---------------------------------------------------------------------------
// FC: C(R x 256) = X(R x 1024) @ W(1024 x 256) + bias, fp32 WMMA 16x16x4.
// Block: 256 threads (8 waves), tile 128 x 128, BK = 32.
// Wave tile: 32 rows x 64 cols -> 8 accumulators, 8 WMMA per K-step.
// ---------------------------------------------------------------------------
__global__ __launch_bounds__(256) void fc_wmma(const float* __restrict__ X,
                                               const float* __restrict__ W,
                                               const float* __restrict__ bias,
                                               float* __restrict__ C) {
  __shared__ __align__(16) float As[128 * 36];  // [m][k], pad 4
  __shared__ __align__(16) float Bs[32 * 132];  // [k][n], pad 4

  const int t    = threadIdx.x;
  const int lane = t & 31;
  const int wave = t >> 5;
  const int wr   = wave & 3;   // 4 row-groups of 32
  const int wc   = wave >> 2;  // 2 col-groups of 64
  const int kh   = lane >> 4;  // K-half select (ISA A/B layout)
  const int nl   = lane & 15;
  const int rowbase = blockIdx.x * 128;
  const int nb      = blockIdx.y * 128;

  v8f acc[2][4];
  for (int mi = 0; mi < 2; ++mi)
    for (int ni = 0; ni < 4; ++ni)
      for (int r = 0; r < 8; ++r) acc[mi][ni][r] = 0.0f;

  const int arow = t >> 1, ah = t & 1;  // A staging: 128 rows x 2 halves of 16
  const int bk = t >> 3, bcg = t & 7;   // B staging: 32 rows x 8 groups at stride 16, 8 floats each

  for (int kb = 0; kb < DHID; kb += 32) {
    __syncthreads();
    {
      const float* src = X + (size_t)(rowbase + arow) * DHID + kb + ah * 16;
      float* dst = &As[arow * 36 + ah * 16];
#pragma unroll
      for (int j = 0; j < 4; ++j)
        *reinterpret_cast<float4*>(dst + 4 * j) =
            *reinterpret_cast<const float4*>(src + 4 * j);
    }
    {
      const float* src = W + (size_t)(kb + bk) * MHID + nb + bcg * 16;
      float* dst = &Bs[bk * 132 + bcg * 16];
#pragma unroll
      for (int j = 0; j < 2; ++j)
        *reinterpret_cast<float4*>(dst + 4 * j) =
            *reinterpret_cast<const float4*>(src + 4 * j);
    }
    if (kb + 32 < DHID) {
      __builtin_prefetch(X + (size_t)(rowbase + arow) * DHID + kb + 32 + ah * 16, 0, 1);
      __builtin_prefetch(W + (size_t)(kb + 32 + bk) * MHID + nb + bcg * 16, 0, 1);
    }
    __syncthreads();
#pragma unroll
    for (int kk = 0; kk < 32; kk += 4) {
      v2f a[2];
#pragma unroll
      for (int mi = 0; mi < 2; ++mi) {
        a[mi][0] = As[(wr * 32 + mi * 16 + nl) * 36 + kk + 2 * kh];
        a[mi][1] = As[(wr * 32 + mi * 16 + nl) * 36 + kk + 2 * kh + 1];
      }
      v2f b[4];
#pragma unroll
      for (int ni = 0; ni < 4; ++ni) {
        b[ni][0] = Bs[(kk + 2 * kh) * 132 + wc * 64 + ni * 16 + nl];
        b[ni][1] = Bs[(kk + 2 * kh + 1) * 132 + wc * 64 + ni * 16 + nl];
      }
#pragma unroll
      for (int mi = 0; mi < 2; ++mi)
#pragma unroll
        for (int ni = 0; ni < 4; ++ni)
          acc[mi][ni] = __builtin_amdgcn_wmma_f32_16x16x4_f32(
              false, a[mi], false, b[ni], (short)0, acc[mi][ni], false, false);
    }
  }
#pragma unroll
  for (int ni = 0; ni < 4; ++ni) {
    const int col = nb + wc * 64 + ni * 16 + nl;
    const float bv = bias[col];
#pragma unroll
    for (int mi = 0; mi < 2; ++mi)
#pragma unroll
      for (int r = 0; r < 8; ++r) {
        const int row = rowbase + wr * 32 + mi * 16 + kh * 8 + r;
        C[(size_t)row * MHID + col] = acc[mi][ni][r] + bv;
      }
  }
}

// ---------------------------------------------------------------------------
// Distance GEMM + LayerNorm(ddof=1, eps on std) + max/argmax over L.
// Tile: 128 query rows x 128 columns (one support sample s), K = 256.
// Both A and B staged [row][k] -> all WMMA operands are contiguous b64 pairs.
// ---------------------------------------------------------------------------
__global__ __launch_bounds__(256) void dist_ln_max(const float* __restrict__ Qm,
                                                   const float* __restrict__ Sm,
                                                   const float* __restrict__ gamma,
                                                   const float* __restrict__ beta,
                                                   float* __restrict__ dmax_out,
                                                   int* __restrict__ imax_out) {
  __shared__ __align__(16) float smem[128 * 132];  // >= 2*(128*36) staging
  float* Qs = smem;              // 128 x 36  [m][k]
  float* Ss = smem + 128 * 36;   // 128 x 36  [n][k]
  float* Ts = smem;              // 128 x 132 reused after K loop

  const int t    = threadIdx.x;
  const int lane = t & 31;
  const int wave = t >> 5;
  const int wr   = wave & 3;
  const int wc   = wave >> 2;
  const int kh   = lane >> 4;
  const int nl   = lane & 15;
  const int e    = blockIdx.z;
  const int s    = blockIdx.y;
  const int qbase = blockIdx.x * 128;
  const size_t Qrow0 = (size_t)e * QL + qbase;     // query rows
  const size_t Srow0 = (size_t)e * QL + s * LSEQ;  // support rows (S*L == Q*L)

  v8f acc[2][4];
  for (int mi = 0; mi < 2; ++mi)
    for (int ni = 0; ni < 4; ++ni)
      for (int r = 0; r < 8; ++r) acc[mi][ni][r] = 0.0f;

  const int arow = t >> 1, ah = t & 1;  // staging: 128 rows x 2 halves of 16

  for (int kb = 0; kb < MHID; kb += 32) {
    __syncthreads();
    {
      const float* src = Qm + (Qrow0 + arow) * MHID + kb + ah * 16;
      float* dst = &Qs[arow * 36 + ah * 16];
#pragma unroll
      for (int j = 0; j < 4; ++j)
        *reinterpret_cast<float4*>(dst + 4 * j) =
            *reinterpret_cast<const float4*>(src + 4 * j);
    }
    {
      const float* src = Sm + (Srow0 + arow) * MHID + kb + ah * 16;
      float* dst = &Ss[arow * 36 + ah * 16];
#pragma unroll
      for (int j = 0; j < 4; ++j)
        *reinterpret_cast<float4*>(dst + 4 * j) =
            *reinterpret_cast<const float4*>(src + 4 * j);
    }
    __syncthreads();
#pragma unroll
    for (int kk = 0; kk < 32; kk += 4) {
      v2f a[2];
#pragma unroll
      for (int mi = 0; mi < 2; ++mi) {
        a[mi][0] = Qs[(wr * 32 + mi * 16 + nl) * 36 + kk + 2 * kh];
        a[mi][1] = Qs[(wr * 32 + mi * 16 + nl) * 36 + kk + 2 * kh + 1];
      }
      v2f b[4];
#pragma unroll
      for (int ni = 0; ni < 4; ++ni) {
        b[ni][0] = Ss[(wc * 64 + ni * 16 + nl) * 36 + kk + 2 * kh];
        b[ni][1] = Ss[(wc * 64 + ni * 16 + nl) * 36 + kk + 2 * kh + 1];
      }
#pragma unroll
      for (int mi = 0; mi < 2; ++mi)
#pragma unroll
        for (int ni = 0; ni < 4; ++ni)
          acc[mi][ni] = __builtin_amdgcn_wmma_f32_16x16x4_f32(
              false, a[mi], false, b[ni], (short)0, acc[mi][ni], false, false);
    }
  }
  __syncthreads();
#pragma unroll
  for (int mi = 0; mi < 2; ++mi)
#pragma unroll
    for (int ni = 0; ni < 4; ++ni)
#pragma unroll
      for (int r = 0; r < 8; ++r)
        Ts[(wr * 32 + mi * 16 + kh * 8 + r) * 132 + wc * 64 + ni * 16 + nl] =
            acc[mi][ni][r];
  __syncthreads();

  // LayerNorm + max/argmax: 4 lanes per row (quad reduce in wave32), 2 passes
  const int part = t & 3;
#pragma unroll
  for (int rr = 0; rr < 2; ++rr) {
    const int row = (t >> 2) + 64 * rr;
    const float* rp = &Ts[row * 132 + part * 32];
    float sum = 0.0f, ss = 0.0f;
#pragma unroll
    for (int j = 0; j < 32; ++j) { const float x = rp[j]; sum += x; ss += x * x; }
    sum += __shfl_xor(sum, 1, 32); sum += __shfl_xor(sum, 2, 32);
    ss  += __shfl_xor(ss, 1, 32);  ss  += __shfl_xor(ss, 2, 32);
    const float mu = sum * (1.0f / 128.0f);
    const float var = fmaxf((ss - 128.0f * mu * mu) * (1.0f / 127.0f), 0.0f);
    const float rstd = 1.0f / (sqrtf(var) + 1e-6f);

    float best = -3.0e38f; int bidx = 0;
#pragma unroll
    for (int j = 0; j < 32; ++j) {
      const int col = part * 32 + j;
      const float y = gamma[col] * (rp[j] - mu) * rstd + beta[col];
      if (y > best) { best = y; bidx = col; }
    }
#pragma unroll
    for (int off = 1; off <= 2; off <<= 1) {
      const float ov = __shfl_xor(best, off, 32);
      const int oi = __shfl_xor(bidx, off, 32);
      if (ov > best || (ov == best && oi < bidx)) { best = ov; bidx = oi; }
    }
    if (part == 0) {
      const size_t o = ((size_t)e * QL + qbase + row) * SUP + s;
      dmax_out[o] = best;
      imax_out[o] = bidx;
    }
  }
}

// ---------------------------------------------------------------------------
// Combine: dist = dh_max[q,hl,:] + dt_max[q,tl,:], argmax/max over S,
// sigmoid, tag gather, d0 / 1-d0 logits and argmax pred.
// One block per (e, q, hl); thread = tl.
// ---------------------------------------------------------------------------
__global__ __launch_bounds__(128) void combine(const float* __restrict__ dh,
                                               const float* __restrict__ dt,
                                               const int* __restrict__ ih,
                                               const int* __restrict__ it,
                                               const int* __restrict__ tags,
                                               float* __restrict__ logits,
                                               float* __restrict__ pred) {
  __shared__ float hv[SUP];
  __shared__ int hx[SUP];
  const int b = blockIdx.x;
  const int e = b / (QRY * LSEQ);
  const int rem = b % (QRY * LSEQ);
  const int q = rem / LSEQ;
  const int hl = rem % LSEQ;
  const int tl = threadIdx.x;

  const size_t hrow = ((size_t)e * QL + q * LSEQ + hl) * SUP;
  if (tl < SUP) { hv[tl] = dh[hrow + tl]; hx[tl] = ih[hrow + tl]; }
  __syncthreads();

  const size_t trow = ((size_t)e * QL + q * LSEQ + tl) * SUP;
  float best = -3.0e38f; int im = 0;
  for (int s = 0; s < SUP; ++s) {
    const float d = hv[s] + dt[trow + s];
    if (d > best) { best = d; im = s; }
  }
  const float dm = 1.0f / (1.0f + expf(-best));
  const int hp = hx[im];
  const int tp = it[trow + im];
  const float mask =
      (float)tags[((size_t)e * SUP + im) * (LSEQ * LSEQ) + hp * LSEQ + tp];
  const float d0 = dm + mask - 2.0f * dm * mask;
  const size_t ob = ((size_t)e * QRY + q) * (LSEQ * LSEQ) + hl * LSEQ + tl;
  logits[ob * 2] = d0;
  logits[ob * 2 + 1] = 1.0f - d0;
  pred[ob] = ((1.0f - d0) > d0) ? 1.0f : 0.0f;
}

// ---------------------------------------------------------------------------
extern "C" void kernel_launch(void* const* d_in, const int* in_sizes, int n_in,
                              void* d_out, int out_size, void* d_ws, size_t ws_size,
                              hipStream_t stream) {
  const float* sup_emb = (const float*)d_in[0];
  const float* qry_emb = (const float*)d_in[1];
  const int* tags_ent = (const int*)d_in[2];
  const int* tags_hr  = (const int*)d_in[3];
  const float* W_ent = (const float*)d_in[4];
  const float* b_ent = (const float*)d_in[5];
  const float* W_hr  = (const float*)d_in[6];
  const float* b_hr  = (const float*)d_in[7];
  const float* gh = (const float*)d_in[8];
  const float* bh = (const float*)d_in[9];
  const float* gt = (const float*)d_in[10];
  const float* bt = (const float*)d_in[11];

  const size_t SZ = (size_t)RSUP * MHID;    // one mapped buffer (floats)
  const size_t MX = (size_t)EP * QL * SUP;  // max/argmax buffer (elems)
  float* ws = (float*)d_ws;
  float* m_s0 = ws;
  float* m_s1 = ws + SZ;
  float* m_q0 = ws + 2 * SZ;
  float* m_q1 = ws + 3 * SZ;
  float* dhmax = ws + 4 * SZ;
  float* dtmax = dhmax + MX;
  int* ih = (int*)(dtmax + MX);
  int* it = ih + MX;

  const size_t LOGN  = (size_t)EP * QRY * LSEQ * LSEQ * 2;  // 5,242,880
  const size_t PREDN = (size_t)EP * QRY * LSEQ * LSEQ;      // 2,621,440
  float* out = (float*)d_out;

  const dim3 gemmGrid(RSUP / 128, MHID / 128);
  const dim3 distGrid(QL / 128, SUP, EP);

  for (int br = 0; br < 2; ++br) {
    const float* W  = br ? W_hr : W_ent;
    const float* bb = br ? b_hr : b_ent;
    const int* tg   = br ? tags_hr : tags_ent;

    fc_wmma<<<gemmGrid, 256, 0, stream>>>(sup_emb, W,               bb,        m_s0);
    fc_wmma<<<gemmGrid, 256, 0, stream>>>(sup_emb, W + DHID * MHID, bb + MHID, m_s1);
    fc_wmma<<<gemmGrid, 256, 0, stream>>>(qry_emb, W,               bb,        m_q0);
    fc_wmma<<<gemmGrid, 256, 0, stream>>>(qry_emb, W + DHID * MHID, bb + MHID, m_q1);

    dist_ln_max<<<distGrid, 256, 0, stream>>>(m_q0, m_s0, gh, bh, dhmax, ih);
    dist_ln_max<<<distGrid, 256, 0, stream>>>(m_q1, m_s1, gt, bt, dtmax, it);

    combine<<<EP * QRY * LSEQ, 128, 0, stream>>>(
        dhmax, dtmax, ih, it, tg,
        out + br * LOGN, out + 2 * LOGN + br * PREDN);
  }
}